// TopKReadout_29377576305109
// MI455X (gfx1250) — compile-verified
//
#include <hip/hip_runtime.h>
#include <hip/hip_bf16.h>

// Problem constants (fixed by the reference).
#define N_ROWS   512
#define S_LEN    2048
#define D_DIM    128
#define CLASSES  1000
#define TOP_K_N  32

typedef __attribute__((ext_vector_type(2))) float v2f;
typedef __attribute__((ext_vector_type(8))) float v8f;

// ---------------------------------------------------------------------------
// Kernel A: per-row  logits -> top-32 -> softmax -> weights row -> summary.
// One block per n. 256 threads = 8 wave32 waves.
// K stream: per key row, each lane loads one float4 (lanes 0..31 cover 128
// floats) => one fully coalesced 512B transaction per wave, 1 MB per block,
// 512 MB total: this is the bandwidth-critical path (~22us at 23.3 TB/s).
// ---------------------------------------------------------------------------
__global__ __launch_bounds__(256) void topk_attn_kernel(
    const float* __restrict__ q,     // [N, D]
    const float* __restrict__ K,     // [N, S, D]
    const float* __restrict__ V,     // [N, S, D]
    float* __restrict__ weights,     // [N, S]  (dense, zero outside top-k)
    float* __restrict__ summary)     // [N, D]  (workspace)
{
    __shared__ float s_logits[S_LEN];                 // 8 KB
    __shared__ unsigned long long s_best;
    __shared__ float s_tv[TOP_K_N];
    __shared__ int   s_ti[TOP_K_N];
    __shared__ float s_w[TOP_K_N];

    const int n    = blockIdx.x;
    const int tid  = threadIdx.x;
    const int lane = tid & 31;
    const int wave = tid >> 5;

    const float* __restrict__ Kn = K + (size_t)n * S_LEN * D_DIM;
    const float* __restrict__ Vn = V + (size_t)n * S_LEN * D_DIM;

    // Each lane holds 4 consecutive floats of q[n].
    const float4 qf = reinterpret_cast<const float4*>(q + (size_t)n * D_DIM)[lane];

    const float inv_sqrt_d = 0.08838834764831845f;    // 1/sqrt(128)

    // ---- logits: each wave streams a contiguous chunk of 256 key rows ----
    const int s_begin = wave * (S_LEN / 8);
    const int s_end   = s_begin + (S_LEN / 8);
    for (int s = s_begin; s < s_end; ++s) {
        const float4 k4 = reinterpret_cast<const float4*>(Kn + (size_t)s * D_DIM)[lane];
        // prefetch a few rows ahead on the streaming path (global_prefetch_b8)
        if (s + 16 < S_LEN)
            __builtin_prefetch(Kn + (size_t)(s + 16) * D_DIM + lane * 4, 0, 0);
        float p = qf.x * k4.x + qf.y * k4.y + qf.z * k4.z + qf.w * k4.w;
        // wave32 reduction
        #pragma unroll
        for (int off = 16; off >= 1; off >>= 1) p += __shfl_xor(p, off);
        if (lane == 0) s_logits[s] = p * inv_sqrt_d;
    }
    __syncthreads();

    // ---- top-32 by iterative argmax extraction (ordered-bits u64 key; ----
    // ---- tie breaks to the smaller index, matching jax.lax.top_k)     ----
    for (int k = 0; k < TOP_K_N; ++k) {
        if (tid == 0) s_best = 0ull;
        __syncthreads();
        unsigned long long local = 0ull;
        for (int i = tid; i < S_LEN; i += 256) {
            unsigned int b = __float_as_uint(s_logits[i]);
            b = (b & 0x80000000u) ? ~b : (b | 0x80000000u);   // monotone map
            unsigned long long key =
                ((unsigned long long)b << 32) | (unsigned int)(S_LEN - 1 - i);
            local = local > key ? local : key;
        }
        atomicMax(&s_best, local);
        __syncthreads();
        if (tid == 0) {
            const unsigned long long best = s_best;
            const int idx = (S_LEN - 1) - (int)(best & 0xFFFFFFFFull);
            unsigned int b = (unsigned int)(best >> 32);
            b = (b & 0x80000000u) ? (b ^ 0x80000000u) : ~b;   // inverse map
            s_tv[k] = __uint_as_float(b);
            s_ti[k] = idx;
            s_logits[idx] = -INFINITY;                        // remove winner
        }
        __syncthreads();
    }

    // ---- softmax over the 32 kept logits (wave 0) ----
    if (tid < TOP_K_N) {
        const float m = s_tv[0];                 // extracted in descending order
        float e = __expf(s_tv[tid] - m);
        float sum = e;
        #pragma unroll
        for (int off = 16; off >= 1; off >>= 1) sum += __shfl_xor(sum, off);
        s_w[tid] = e / sum;
    }
    __syncthreads();

    // ---- dense weights row: zeros then scatter top-k ----
    float* __restrict__ wrow = weights + (size_t)n * S_LEN;
    for (int i = tid; i < S_LEN; i += 256) wrow[i] = 0.0f;
    __syncthreads();
    if (tid < TOP_K_N) wrow[s_ti[tid]] = s_w[tid];

    // ---- summary[n, d] = sum_j w_j * V[n, idx_j, d]  (gather 32 rows) ----
    if (tid < D_DIM) {
        float acc = 0.0f;
        #pragma unroll 8
        for (int j = 0; j < TOP_K_N; ++j)
            acc += s_w[j] * Vn[(size_t)s_ti[j] * D_DIM + tid];
        summary[(size_t)n * D_DIM + tid] = acc;
    }
}

// ---------------------------------------------------------------------------
// Kernel B: out[M, outDim] = summary[M,128] @ W[outDim,128]^T + bias
// Exact fp32 path via V_WMMA_F32_16X16X4_F32: one wave computes one 16x16
// output tile, accumulating 32 WMMAs over K=128 in steps of 4.
// A frag (16x4 f32): lanes 0-15 hold M=lane, K=k..k+1; lanes 16-31 K=k+2..k+3.
// B frag (4x16 f32): mirrored with N=lane%16.
// C/D (16x16 f32): VGPR r holds M = r + 8*(lane>=16), N = lane%16.
// ---------------------------------------------------------------------------
__global__ __launch_bounds__(32) void readout_gemm_wmma(
    const float* __restrict__ Ssum,   // [N_ROWS, 128]
    const float* __restrict__ W,      // [outDim, 128]
    const float* __restrict__ bias,   // [outDim]
    float* __restrict__ out,          // [N_ROWS, outDim]
    int outDim)
{
    const int lane = threadIdx.x;        // 0..31, wave32
    const int half = lane >> 4;          // 0: K lo pair, 1: K hi pair
    const int l    = lane & 15;
    const int m0   = blockIdx.y * 16;
    const int n0   = blockIdx.x * 16;

    const int   col  = n0 + l;
    const int   ccol = col < outDim ? col : outDim - 1;   // clamped, masked
    const float mask = col < outDim ? 1.0f : 0.0f;

    const float* __restrict__ arow = Ssum + (size_t)(m0 + l) * D_DIM;
    const float* __restrict__ brow = W    + (size_t)ccol     * D_DIM;

    v8f c = {0.f, 0.f, 0.f, 0.f, 0.f, 0.f, 0.f, 0.f};

    #pragma unroll 4
    for (int k = 0; k < D_DIM; k += 4) {
        v2f a, b;
        a.x = arow[k + 2 * half];
        a.y = arow[k + 2 * half + 1];
        b.x = brow[k + 2 * half] * mask;
        b.y = brow[k + 2 * half + 1] * mask;
        // (neg_a, A, neg_b, B, c_mod, C, reuse_a, reuse_b)
        c = __builtin_amdgcn_wmma_f32_16x16x4_f32(
                false, a, false, b, (short)0, c, false, false);
    }

    if (col < outDim) {
        const float bv = bias[col];
        #pragma unroll
        for (int r = 0; r < 8; ++r) {
            const int row = m0 + r + half * 8;
            out[(size_t)row * outDim + col] = c[r] + bv;
        }
    }
}

// ---------------------------------------------------------------------------
extern "C" void kernel_launch(void* const* d_in, const int* in_sizes, int n_in,
                              void* d_out, int out_size, void* d_ws, size_t ws_size,
                              hipStream_t stream) {
    const float* q     = (const float*)d_in[0];   // [512,128]
    const float* K     = (const float*)d_in[1];   // [512,2048,128]
    const float* V     = (const float*)d_in[2];   // [512,2048,128]
    const float* W_cls = (const float*)d_in[3];   // [1000,128]
    const float* b_cls = (const float*)d_in[4];   // [1000]
    const float* W_rec = (const float*)d_in[5];   // [128,128]
    const float* b_rec = (const float*)d_in[6];   // [128]

    float* out     = (float*)d_out;
    float* cls_out = out;                                    // [512,1000]
    float* rec_out = cls_out + (size_t)N_ROWS * CLASSES;     // [512,128]
    float* wts_out = rec_out + (size_t)N_ROWS * D_DIM;       // [512,2048]

    float* summary = (float*)d_ws;                           // [512,128]

    // Kernel A: bandwidth-bound K stream + top-k + softmax + gather.
    topk_attn_kernel<<<dim3(N_ROWS), dim3(256), 0, stream>>>(
        q, K, V, wts_out, summary);

    // Kernel B: fp32 WMMA readout GEMMs.
    readout_gemm_wmma<<<dim3((CLASSES + 15) / 16, N_ROWS / 16), dim3(32), 0, stream>>>(
        summary, W_cls, b_cls, cls_out, CLASSES);
    readout_gemm_wmma<<<dim3(D_DIM / 16, N_ROWS / 16), dim3(32), 0, stream>>>(
        summary, W_rec, b_rec, rec_out, D_DIM);
}